// RelativePositionEncoding_89404039233854
// MI455X (gfx1250) — compile-verified
//
#include <hip/hip_runtime.h>

// ---- problem constants (from the reference) ----
#define R_MAX   32
#define S_MAX   2
#define CZ      128
#define NRES    (2 * R_MAX + 2)                     // 66
#define NCHAIN  (2 * S_MAX + 2)                     // 6
#define FDIM    (2 * NRES + 1 + NCHAIN)             // 139
#define WELEMS  (FDIM * CZ)                         // 17792 floats
#define WBYTES  (WELEMS * 4)                        // 71168 bytes (fits 320KB/WGP LDS)

// special rows of W (see analysis)
#define ROW_RES_INTER   (2 * R_MAX + 1)             // 65  : d_res when chains differ
#define ROW_TOK_INTER   (NRES + 2 * R_MAX + 1)      // 131 : d_tok unless (same chain & same residue)
#define ROW_ENT         (2 * NRES)                  // 132 : same_entity row
#define ROW_CHN_BASE    (2 * NRES + 1)              // 133 : chain rows base
#define ROW_CHN_SAME    (ROW_CHN_BASE + 2 * S_MAX + 1) // 138 : d_chain when same chain

typedef float v4f __attribute__((ext_vector_type(4)));

__device__ __forceinline__ int iclamp(int x, int lo, int hi) {
  x = x < lo ? lo : x;
  return x > hi ? hi : x;
}

__global__ __launch_bounds__(256)
void relpos_encode_kernel(const int* __restrict__ asym,
                          const int* __restrict__ resi,
                          const int* __restrict__ ent,
                          const int* __restrict__ tok,
                          const int* __restrict__ sym,
                          const float* __restrict__ W,
                          float* __restrict__ out,
                          int N)
{
  // The only LDS object in this kernel -> placed at LDS byte offset 0,
  // which the async-copy inline asm below relies on.
  __shared__ __align__(16) float lds_w[WELEMS];

  const int tid = threadIdx.x;

  // ---------------------------------------------------------------
  // Stage W (139x128 fp32 = 71,168 B) into LDS with CDNA5 async
  // global->LDS copies: 16 B per lane per op, tracked by ASYNCcnt.
  //
  // IMPORTANT: lds_w's address is passed as an (unused) asm operand so
  // the LDS object *escapes*. Otherwise LLVM proves the "memory"
  // clobber can't touch it, treats lds_w as never-written, and folds
  // every ds_load from it to undef (observed in round 0: the whole
  // compute loop collapsed to storing NaN).
  // ---------------------------------------------------------------
  {
    const char* wb = (const char*)W;
    float* lds_escape = lds_w;               // escapes into the asm below
    for (int off = tid * 16; off < WBYTES; off += 256 * 16) {
      unsigned lds_off = (unsigned)off;      // LDS byte address (array at offset 0)
      const char* g = wb + off;              // 64-bit global address (GV mode)
      asm volatile("global_load_async_to_lds_b128 %0, %1, off"
                   :: "v"(lds_off), "v"(g), "v"(lds_escape)
                   : "memory");
    }
    asm volatile("s_wait_asynccnt 0x0" ::: "memory");
  }
  __syncthreads();   // make every wave's async-filled LDS visible to all waves

  // ---------------------------------------------------------------
  // Mapping: block = row j, wave = one (j,i) pair per iteration,
  // lane l = channels 4l..4l+3  ->  one ds_load_b128 gather and one
  // fully-coalesced 512B non-temporal store per pair.
  // ---------------------------------------------------------------
  const int lane = tid & 31;
  const int wave = tid >> 5;               // 0..7
  const int j    = blockIdx.x;             // row index (first N axis of p)
  const int cb   = lane * 4;               // channel base for this lane

  const int aj = asym[j], rj = resi[j], ej = ent[j], tj = tok[j], sj = sym[j];

  const float* lw = lds_w;
  // Register-resident fixed rows (4 floats per lane each):
  const v4f w_ent      = *(const v4f*)(lw + ROW_ENT       * CZ + cb);
  const v4f w_chn_same = *(const v4f*)(lw + ROW_CHN_SAME  * CZ + cb);
  const v4f w_tok_int  = *(const v4f*)(lw + ROW_TOK_INTER * CZ + cb);
  const v4f w_inter    = *(const v4f*)(lw + ROW_RES_INTER * CZ + cb) + w_tok_int;

  for (int i = wave; i < N; i += 8) {
    const int ai = asym[i];
    const int ei = ent[i];
    v4f acc;
    if (ai == aj) {                                       // wave-uniform branch
      const int ri   = resi[i];
      const int dres = iclamp(ri - rj + R_MAX, 0, 2 * R_MAX);
      acc = *(const v4f*)(lw + dres * CZ + cb);           // 1 LDS gather
      if (ri == rj) {                                     // diagonal only
        const int ti   = tok[i];
        const int dtok = iclamp(ti - tj + R_MAX, 0, 2 * R_MAX);
        acc += *(const v4f*)(lw + (NRES + dtok) * CZ + cb);
      } else {
        acc += w_tok_int;                                 // fixed row, registers
      }
      acc += w_chn_same;                                  // fixed row, registers
    } else {
      const int si   = sym[i];
      const int dchn = iclamp(si - sj + S_MAX, 0, 2 * S_MAX);
      acc = w_inter + *(const v4f*)(lw + (ROW_CHN_BASE + dchn) * CZ + cb); // 1 LDS gather
    }
    const float e = (ei == ej) ? 1.0f : 0.0f;
    acc += e * w_ent;                                     // v_fma per component

    // Streaming 512MB output >> 192MB L2: non-temporal 128-bit store.
    __builtin_nontemporal_store(acc, (v4f*)(out + ((size_t)j * N + i) * CZ + cb));
  }
}

extern "C" void kernel_launch(void* const* d_in, const int* in_sizes, int n_in,
                              void* d_out, int out_size, void* d_ws, size_t ws_size,
                              hipStream_t stream) {
  (void)n_in; (void)out_size; (void)d_ws; (void)ws_size;
  const int*   asym = (const int*)d_in[0];   // asym_id        (int32)
  const int*   resi = (const int*)d_in[1];   // residue_index  (int32)
  const int*   ent  = (const int*)d_in[2];   // entity_id      (int32)
  const int*   tok  = (const int*)d_in[3];   // token_index    (int32)
  const int*   sym  = (const int*)d_in[4];   // sym_id         (int32)
  const float* W    = (const float*)d_in[5]; // (139, 128) fp32

  float* out = (float*)d_out;                // (1, N, N, 128) fp32
  const int N = in_sizes[0];                 // B == 1 -> N tokens

  relpos_encode_kernel<<<N, 256, 0, stream>>>(asym, resi, ent, tok, sym, W, out, N);
}